// GConvLayerEdges_28071906247357
// MI455X (gfx1250) — compile-verified
//
#include <hip/hip_runtime.h>

#define NN 40000
#define NE 640000
#define DD 128
#define EPS 1e-5f

typedef __attribute__((ext_vector_type(2))) float v2f;
typedef __attribute__((ext_vector_type(8))) float v8f;

// ---------------------------------------------------------------------------
// Kernel 1: y = relu(x @ W^T + b)   (N x 128) = (N x 128)(128 x 128)
// One wave per 16x16 output tile via V_WMMA_F32_16X16X4_F32 (fp32-exact).
// Block = 256 threads (8 waves): wave w covers column tile w (16 cols),
// all waves share the same 16-row A slab (L0/L2 reuse).
// ---------------------------------------------------------------------------
__global__ __launch_bounds__(256) void gemm_relu_wmma(
    const float* __restrict__ x, const float* __restrict__ W,
    const float* __restrict__ bias, float* __restrict__ y)
{
    const int wave    = threadIdx.x >> 5;     // 0..7 -> column tile
    const int lane    = threadIdx.x & 31;
    const int rowBase = blockIdx.x << 4;      // 16 rows per block
    const int colBase = wave << 4;            // 16 cols per wave
    const int m       = lane & 15;            // M index (A) / N index (B,C,D)
    const int half    = lane >> 4;            // 0: K 0..1, 1: K 2..3 per step

    const float* xrow = x + (size_t)(rowBase + m) * DD;   // A row (M = m)
    const float* wrow = W + (size_t)(colBase + m) * DD;   // W row j == col j of W^T

    v8f c = {};
    #pragma unroll
    for (int k = 0; k < DD; k += 4) {
        // A 16x4 fp32: lanes 0-15 hold K={0,1}, lanes 16-31 hold K={2,3}
        const float2 av = *(const float2*)(xrow + k + 2 * half);
        // B 4x16 fp32 (= W^T tile): lane holds column N=m, same K split
        const float2 bv = *(const float2*)(wrow + k + 2 * half);
        v2f a; a.x = av.x; a.y = av.y;
        v2f b; b.x = bv.x; b.y = bv.y;
        c = __builtin_amdgcn_wmma_f32_16x16x4_f32(
                /*neg_a=*/false, a, /*neg_b=*/false, b,
                /*c_mod=*/(short)0, c, /*reuse_a=*/false, /*reuse_b=*/false);
    }

    const float bj = bias[colBase + m];       // per-column bias (N = m)
    #pragma unroll
    for (int v = 0; v < 8; ++v) {
        const int row = rowBase + v + 8 * half;      // C/D: M = v + 8*half
        float val = c[v] + bj;
        val = val > 0.f ? val : 0.f;                 // relu
        y[(size_t)row * DD + colBase + m] = val;
    }
}

// ---------------------------------------------------------------------------
// Kernel 0: zero agg (float 0.0 == bit pattern 0 == u32 atomic-max identity
// for non-negative floats)
// ---------------------------------------------------------------------------
__global__ __launch_bounds__(256) void zero_agg(float4* __restrict__ agg, int n4)
{
    int i = blockIdx.x * blockDim.x + threadIdx.x;
    if (i < n4) agg[i] = make_float4(0.f, 0.f, 0.f, 0.f);
}

// ---------------------------------------------------------------------------
// Kernel 2: segment_max scatter. One wave per edge; lane handles 4 floats.
// relu output >= 0, so uint bit-pattern max == float max -> single
// GLOBAL_ATOMIC_MAX_U32 per element (no CAS loop).
// ---------------------------------------------------------------------------
__global__ __launch_bounds__(256) void scatter_max(
    const int* __restrict__ e, const float* __restrict__ y,
    unsigned int* __restrict__ agg)
{
    const int gid  = blockIdx.x * blockDim.x + threadIdx.x;
    const int edge = gid >> 5;
    const int lane = gid & 31;
    if (edge >= NE) return;

    const int2 sd = *(const int2*)(e + (size_t)edge * 2);  // (src, dst)
    const float4 v = *(const float4*)(y + (size_t)sd.x * DD + lane * 4);
    unsigned int* a = agg + (size_t)sd.y * DD + lane * 4;
    atomicMax(a + 0, __float_as_uint(v.x));
    atomicMax(a + 1, __float_as_uint(v.y));
    atomicMax(a + 2, __float_as_uint(v.z));
    atomicMax(a + 3, __float_as_uint(v.w));
}

// ---------------------------------------------------------------------------
// Kernel 3: h = x + agg; out = h * rsqrt(mean(h^2)+eps) * g + rb
// One wave32 per row (32 lanes x float4 = 128), shfl_xor reduction.
// ---------------------------------------------------------------------------
__global__ __launch_bounds__(256) void residual_rmsnorm(
    const float* __restrict__ x, const float* __restrict__ agg,
    const float* __restrict__ g, const float* __restrict__ rb,
    float* __restrict__ out)
{
    const int wave = threadIdx.x >> 5;
    const int lane = threadIdx.x & 31;
    const int row  = blockIdx.x * 8 + wave;          // 40000 % 8 == 0
    if (row >= NN) return;

    const size_t base = (size_t)row * DD + lane * 4;
    const float4 xv = *(const float4*)(x + base);
    const float4 av = *(const float4*)(agg + base);
    float4 h;
    h.x = xv.x + av.x; h.y = xv.y + av.y;
    h.z = xv.z + av.z; h.w = xv.w + av.w;

    float ss = h.x * h.x + h.y * h.y + h.z * h.z + h.w * h.w;
    #pragma unroll
    for (int off = 16; off > 0; off >>= 1)
        ss += __shfl_xor(ss, off, 32);

    const float inv = rsqrtf(ss * (1.0f / DD) + EPS);

    const float4 gv = *(const float4*)(g  + lane * 4);
    const float4 rv = *(const float4*)(rb + lane * 4);
    float4 o;
    o.x = h.x * inv * gv.x + rv.x;
    o.y = h.y * inv * gv.y + rv.y;
    o.z = h.z * inv * gv.z + rv.z;
    o.w = h.w * inv * gv.w + rv.w;
    *(float4*)(out + base) = o;
}

// ---------------------------------------------------------------------------
extern "C" void kernel_launch(void* const* d_in, const int* in_sizes, int n_in,
                              void* d_out, int out_size, void* d_ws, size_t ws_size,
                              hipStream_t stream) {
    const float* x  = (const float*)d_in[0];
    const int*   e  = (const int*)  d_in[1];
    const float* W  = (const float*)d_in[2];
    const float* b  = (const float*)d_in[3];
    const float* g  = (const float*)d_in[4];
    const float* rb = (const float*)d_in[5];
    float* out = (float*)d_out;

    float* y   = (float*)d_ws;                      // N*128 floats (20.5 MB)
    float* agg = y + (size_t)NN * DD;               // N*128 floats (20.5 MB)

    // 1) zero the aggregation buffer (atomic-max identity)
    {
        const int n4 = NN * DD / 4;                 // 1,280,000 float4s
        zero_agg<<<(n4 + 255) / 256, 256, 0, stream>>>((float4*)agg, n4);
    }
    // 2) y = relu(x @ W^T + b) via fp32 WMMA
    gemm_relu_wmma<<<NN / 16, 256, 0, stream>>>(x, W, b, y);
    // 3) per-edge scatter max (one wave per edge)
    {
        const long long threads = (long long)NE * 32;
        scatter_max<<<(int)((threads + 255) / 256), 256, 0, stream>>>(
            e, y, (unsigned int*)agg);
    }
    // 4) residual + RMSNorm
    residual_rmsnorm<<<NN / 8, 256, 0, stream>>>(x, agg, g, rb, out);
}